// RegionGCN_39247411151461
// MI455X (gfx1250) — compile-verified
//
#include <hip/hip_runtime.h>
#include <hip/hip_bf16.h>

// Problem constants (from the reference)
#define N_NODES 100000
#define N_EDGES 1600000
#define IN_DIM  512
#define HID_DIM 128
#define OUT_DIM 64
#define BN_EPS  1e-5f

typedef float v2f __attribute__((ext_vector_type(2)));
typedef float v8f __attribute__((ext_vector_type(8)));

// ---------------------------------------------------------------------------
// Zero-fill (needed every launch: harness does not re-zero scratch)
// ---------------------------------------------------------------------------
__global__ __launch_bounds__(256) void zero_f32_kernel(float* __restrict__ p, int n) {
    int i = blockIdx.x * 256 + threadIdx.x;
    if (i < n) p[i] = 0.0f;
}

// ---------------------------------------------------------------------------
// GEMM1: h1[100000x128] = x[100000x512] @ W1[512x128], fp32 WMMA 16x16x4.
// Block = 256 threads (8 waves). Block computes 16 rows x 128 cols.
// Wave w owns the 16x16 tile at column w*16. A-tile staged in LDS with
// padded stride (516) to avoid 64-bank conflicts on stride-512 rows.
// ---------------------------------------------------------------------------
#define A1_STRIDE 516  // 516*4B per row; %16B == 0 so float4 stores stay aligned
__global__ __launch_bounds__(256) void gemm1_wmma_kernel(const float* __restrict__ x,
                                                         const float* __restrict__ W1,
                                                         float* __restrict__ h1) {
    __shared__ __align__(16) float As[16 * A1_STRIDE];  // ~33KB
    const int tid = threadIdx.x;
    const int m0  = blockIdx.x * 16;

    // Cooperative load of the 16x512 A tile (2048 float4 chunks, 8 per thread)
    for (int c = tid; c < 16 * (IN_DIM / 4); c += 256) {
        int row = c >> 7;          // /128 chunks per row
        int q   = c & 127;
        *(float4*)(&As[row * A1_STRIDE + q * 4]) =
            ((const float4*)(x + (size_t)(m0 + row) * IN_DIM))[q];
    }
    __syncthreads();

    const int wave  = tid >> 5;           // 0..7 -> column tile
    const int lane  = tid & 31;
    const int mrow  = lane & 15;          // A row for this lane (both halves)
    const int khalf = (lane >> 4) << 1;   // 0 or 2: K sub-offset per ISA layout
    const int n     = wave * 16 + (lane & 15);

    v8f acc = {};
    for (int k0 = 0; k0 < IN_DIM; k0 += 4) {
        v2f a, b;
        // A 16x4 fp32 layout: VGPR0=K(khalf), VGPR1=K(khalf+1)
        a.x = As[mrow * A1_STRIDE + k0 + khalf];
        a.y = As[mrow * A1_STRIDE + k0 + khalf + 1];
        // B 4x16 fp32 layout: row K striped across lanes within a VGPR
        b.x = W1[(size_t)(k0 + khalf) * HID_DIM + n];
        b.y = W1[(size_t)(k0 + khalf + 1) * HID_DIM + n];
        // keep B slice warm (emits global_prefetch_b8); (k0+32)&511 is always in-bounds
        __builtin_prefetch(&W1[(size_t)((k0 + 32) & (IN_DIM - 1)) * HID_DIM + n], 0, 1);
        acc = __builtin_amdgcn_wmma_f32_16x16x4_f32(false, a, false, b,
                                                    (short)0, acc, false, false);
    }

    // C/D layout: VGPR v -> row v (lanes 0-15) / row v+8 (lanes 16-31), col = lane%16
    const int rbase = m0 + ((lane >> 4) << 3);
#pragma unroll
    for (int v = 0; v < 8; ++v)
        h1[(size_t)(rbase + v) * HID_DIM + n] = acc[v];
}

// ---------------------------------------------------------------------------
// Edge scatter-add, layer 1: agg[dst] += h[src], 128 channels.
// One wave per edge: lane g handles channels [4g, 4g+4). Gather is a
// coalesced float4; scatter is 4 global_atomic_add_f32 (L2-resident).
// ---------------------------------------------------------------------------
__global__ __launch_bounds__(256) void scatter1_kernel(const float* __restrict__ h,
                                                       const int* __restrict__ src,
                                                       const int* __restrict__ dst,
                                                       float* __restrict__ agg) {
    unsigned long long t = (unsigned long long)blockIdx.x * 256ull + threadIdx.x;
    unsigned int e = (unsigned int)(t >> 5);      // edge
    unsigned int g = (unsigned int)(t & 31);      // 4-channel group
    if (e >= N_EDGES) return;
    int s = src[e], d = dst[e];
    float4 v = *(const float4*)(h + (size_t)s * HID_DIM + g * 4);
    float* ap = agg + (size_t)d * HID_DIM + g * 4;
    atomicAdd(ap + 0, v.x);
    atomicAdd(ap + 1, v.y);
    atomicAdd(ap + 2, v.z);
    atomicAdd(ap + 3, v.w);
}

// ---------------------------------------------------------------------------
// BN stats: per-channel sum / sumsq over all nodes. 128 threads = 128
// channels; each block strides 250 rows, then one atomicAdd per channel.
// (b1 is omitted: a per-channel constant shift cancels exactly in BN.)
// ---------------------------------------------------------------------------
#define BN_ROWS 250
__global__ __launch_bounds__(128) void bn_stats_kernel(const float* __restrict__ agg,
                                                       float* __restrict__ stats) {
    const int c  = threadIdx.x;
    const int r0 = blockIdx.x * BN_ROWS;
    float s = 0.f, s2 = 0.f;
    for (int r = r0; r < r0 + BN_ROWS; ++r) {
        float v = agg[(size_t)r * HID_DIM + c];
        s += v; s2 += v * v;
    }
    atomicAdd(&stats[c], s);
    atomicAdd(&stats[HID_DIM + c], s2);
}

// ---------------------------------------------------------------------------
// BN apply + ReLU, in place on agg1.
// ---------------------------------------------------------------------------
__global__ __launch_bounds__(256) void bn_apply_relu_kernel(float* __restrict__ agg,
                                                            const float* __restrict__ stats,
                                                            const float* __restrict__ gamma,
                                                            const float* __restrict__ beta) {
    int i = blockIdx.x * 256 + threadIdx.x;
    if (i >= N_NODES * HID_DIM) return;
    int c = i & (HID_DIM - 1);
    const float invN = 1.0f / (float)N_NODES;
    float mean = stats[c] * invN;
    float var  = stats[HID_DIM + c] * invN - mean * mean;
    float sc   = rsqrtf(var + BN_EPS) * gamma[c];
    float v    = (agg[i] - mean) * sc + beta[c];
    agg[i] = fmaxf(v, 0.0f);
}

// ---------------------------------------------------------------------------
// GEMM2: h2[100000x64] = hbn[100000x128] @ W2[128x64], fp32 WMMA 16x16x4.
// Block = 256 threads (8 waves) -> 32 rows x 64 cols (2x4 wave tiles).
// ---------------------------------------------------------------------------
#define A2_STRIDE 132  // 132*4B per row; %16B == 0
__global__ __launch_bounds__(256) void gemm2_wmma_kernel(const float* __restrict__ hbn,
                                                         const float* __restrict__ W2,
                                                         float* __restrict__ h2) {
    __shared__ __align__(16) float As[32 * A2_STRIDE];  // ~16.9KB
    const int tid = threadIdx.x;
    const int m0  = blockIdx.x * 32;

    for (int c = tid; c < 32 * (HID_DIM / 4); c += 256) {
        int row = c >> 5;          // /32 chunks per row
        int q   = c & 31;
        *(float4*)(&As[row * A2_STRIDE + q * 4]) =
            ((const float4*)(hbn + (size_t)(m0 + row) * HID_DIM))[q];
    }
    __syncthreads();

    const int wave  = tid >> 5;
    const int lane  = tid & 31;
    const int mt    = wave >> 2;          // 0..1 row tile
    const int nt    = wave & 3;           // 0..3 col tile
    const int mrow  = mt * 16 + (lane & 15);
    const int khalf = (lane >> 4) << 1;
    const int n     = nt * 16 + (lane & 15);

    v8f acc = {};
    for (int k0 = 0; k0 < HID_DIM; k0 += 4) {
        v2f a, b;
        a.x = As[mrow * A2_STRIDE + k0 + khalf];
        a.y = As[mrow * A2_STRIDE + k0 + khalf + 1];
        b.x = W2[(size_t)(k0 + khalf) * OUT_DIM + n];
        b.y = W2[(size_t)(k0 + khalf + 1) * OUT_DIM + n];
        __builtin_prefetch(&W2[(size_t)((k0 + 32) & (HID_DIM - 1)) * OUT_DIM + n], 0, 1);
        acc = __builtin_amdgcn_wmma_f32_16x16x4_f32(false, a, false, b,
                                                    (short)0, acc, false, false);
    }

    const int rbase = m0 + mt * 16 + ((lane >> 4) << 3);
#pragma unroll
    for (int v = 0; v < 8; ++v)
        h2[(size_t)(rbase + v) * OUT_DIM + n] = acc[v];
}

// ---------------------------------------------------------------------------
// Edge scatter-add, layer 2: 64 channels; lanes 0..15 of each half-wave
// serve one edge (16 groups of 4 channels).
// ---------------------------------------------------------------------------
__global__ __launch_bounds__(256) void scatter2_kernel(const float* __restrict__ h,
                                                       const int* __restrict__ src,
                                                       const int* __restrict__ dst,
                                                       float* __restrict__ agg) {
    unsigned long long t = (unsigned long long)blockIdx.x * 256ull + threadIdx.x;
    unsigned int e = (unsigned int)(t >> 4);
    unsigned int g = (unsigned int)(t & 15);
    if (e >= N_EDGES) return;
    int s = src[e], d = dst[e];
    float4 v = *(const float4*)(h + (size_t)s * OUT_DIM + g * 4);
    float* ap = agg + (size_t)d * OUT_DIM + g * 4;
    atomicAdd(ap + 0, v.x);
    atomicAdd(ap + 1, v.y);
    atomicAdd(ap + 2, v.z);
    atomicAdd(ap + 3, v.w);
}

// ---------------------------------------------------------------------------
// Row-wise L2 normalize: one wave32 per node, 2 channels per lane,
// shfl_xor butterfly reduction. Adds b2 first (reference: agg + b2).
// ---------------------------------------------------------------------------
__global__ __launch_bounds__(256) void l2norm_kernel(const float* __restrict__ agg,
                                                     const float* __restrict__ b2,
                                                     float* __restrict__ out) {
    int lane = threadIdx.x & 31;
    int node = (blockIdx.x * 256 + threadIdx.x) >> 5;
    if (node >= N_NODES) return;
    float v0 = agg[(size_t)node * OUT_DIM + lane]      + b2[lane];
    float v1 = agg[(size_t)node * OUT_DIM + 32 + lane] + b2[32 + lane];
    float ss = v0 * v0 + v1 * v1;
#pragma unroll
    for (int off = 16; off > 0; off >>= 1) ss += __shfl_xor(ss, off, 32);
    float inv = 1.0f / fmaxf(sqrtf(ss), 1e-12f);
    out[(size_t)node * OUT_DIM + lane]      = v0 * inv;
    out[(size_t)node * OUT_DIM + 32 + lane] = v1 * inv;
}

// ---------------------------------------------------------------------------
// Launch
// ---------------------------------------------------------------------------
extern "C" void kernel_launch(void* const* d_in, const int* in_sizes, int n_in,
                              void* d_out, int out_size, void* d_ws, size_t ws_size,
                              hipStream_t stream) {
    const float* x     = (const float*)d_in[0];
    const int*   ei    = (const int*)  d_in[1];   // [2, N_EDGES]
    const float* W1    = (const float*)d_in[2];
    // d_in[3] = b1: per-channel constant shift cancels exactly inside BN -> unused
    const float* W2    = (const float*)d_in[4];
    const float* b2    = (const float*)d_in[5];
    const float* gamma = (const float*)d_in[6];
    const float* beta  = (const float*)d_in[7];
    float* out = (float*)d_out;

    const int* src = ei;
    const int* dst = ei + N_EDGES;

    // Workspace layout (floats). h1 region is reused for h2/agg2 once dead.
    float* f     = (float*)d_ws;
    float* h1    = f;                                   // 12.8M  [N, 128]
    float* agg1  = f + (size_t)N_NODES * HID_DIM;       // 12.8M  [N, 128]
    float* stats = agg1 + (size_t)N_NODES * HID_DIM;    // 256 (sum | sumsq)
    float* h2    = f;                                   // 6.4M   [N, 64] (reuse h1 lo)
    float* agg2  = f + (size_t)N_NODES * OUT_DIM;       // 6.4M   [N, 64] (reuse h1 hi)

    const int aggstats = N_NODES * HID_DIM + 2 * HID_DIM;  // agg1 + stats contiguous

    // 1) zero agg1 + BN stats
    zero_f32_kernel<<<(aggstats + 255) / 256, 256, 0, stream>>>(agg1, aggstats);
    // 2) h1 = x @ W1 (fp32 WMMA)
    gemm1_wmma_kernel<<<N_NODES / 16, 256, 0, stream>>>(x, W1, h1);
    // 3) agg1[dst] += h1[src]
    scatter1_kernel<<<(N_EDGES * 32) / 256, 256, 0, stream>>>(h1, src, dst, agg1);
    // 4) BN stats
    bn_stats_kernel<<<N_NODES / BN_ROWS, 128, 0, stream>>>(agg1, stats);
    // 5) BN apply + ReLU in place (agg1 -> hbn)
    bn_apply_relu_kernel<<<(N_NODES * HID_DIM) / 256, 256, 0, stream>>>(agg1, stats, gamma, beta);
    // 6) zero agg2 (h1 now dead)
    zero_f32_kernel<<<(N_NODES * OUT_DIM + 255) / 256, 256, 0, stream>>>(agg2, N_NODES * OUT_DIM);
    // 7) h2 = hbn @ W2 (fp32 WMMA)
    gemm2_wmma_kernel<<<N_NODES / 32, 256, 0, stream>>>(agg1, W2, h2);
    // 8) agg2[dst] += h2[src]
    scatter2_kernel<<<(N_EDGES * 16) / 256, 256, 0, stream>>>(h2, src, dst, agg2);
    // 9) out = l2_normalize(agg2 + b2)
    l2norm_kernel<<<(N_NODES * 32 + 255) / 256, 256, 0, stream>>>(agg2, b2, out);
}